// NaturalLogicPSLLoss_37950331028212
// MI455X (gfx1250) — compile-verified
//
#include <hip/hip_runtime.h>
#include <stdint.h>

// ---------------------------------------------------------------------------
// NaturalLogicPSLLoss on MI455X (gfx1250)
//
//   composed[b,t,:] = (p_t ⊗ p_t1) · S,  S[pair,k] = (comp_table[pair] == k)
//   -> per 16 positions: 13x V_WMMA_F32_16X16X4_F32 over the padded 52-wide
//      pair dimension; the 343-MAC/position contraction rides the WMMA pipe.
//   HBM -> LDS row staging via global_load_async_to_lds_b128 (ASYNCcnt),
//   so the 117 MB input is read from HBM exactly once (~5us roofline).
// ---------------------------------------------------------------------------

#define NREL   7
#define MLEN   512
#define ROWF   (MLEN * NREL)     // 3584 floats per batch row
#define ROWB   (ROWF * 4)        // 14336 bytes
#define CHUNKS (ROWB / 16)       // 896 x 16B async chunks
#define LDSF   3620              // padded: max idx (513*7+15)=3606

typedef float v2f __attribute__((ext_vector_type(2)));
typedef float v8f __attribute__((ext_vector_type(8)));

#define FOR13(M) M(0) M(1) M(2) M(3) M(4) M(5) M(6) M(7) M(8) M(9) M(10) M(11) M(12)

// A-operand entry for flat pair index pr (compile-time): p[pr/7]*q[pr%7], 0 pad
#define PQV(pr) ((pr) < 49 ? pq[(pr) / 7] * pq[7 + ((pr) % 7)] : 0.0f)
// comp_table lookup for pair pr (compile-time index -> uniform s_load), pad->999
#define CPV(pr) ((pr) < 49 ? comp[(pr)] : 999)

// B operand for k-step s: lane L, reg v holds B[2*(L/16)+v][L%16]
#define BDEF(s)                                                              \
  v2f bs##s;                                                                 \
  {                                                                          \
    const int c0 = hsel ? CPV(4 * (s) + 2) : CPV(4 * (s) + 0);               \
    const int c1 = hsel ? CPV(4 * (s) + 3) : CPV(4 * (s) + 1);               \
    bs##s.x = (c0 == m) ? 1.0f : 0.0f;                                       \
    bs##s.y = (c1 == m) ? 1.0f : 0.0f;                                       \
  }

// One K=4 WMMA step: A lane L reg v = O[L%16][4s + 2*(L/16) + v]
#define WSTEP(s)                                                             \
  {                                                                          \
    v2f a;                                                                   \
    a.x = hsel ? PQV(4 * (s) + 2) : PQV(4 * (s) + 0);                        \
    a.y = hsel ? PQV(4 * (s) + 3) : PQV(4 * (s) + 1);                        \
    acc8 = __builtin_amdgcn_wmma_f32_16x16x4_f32(                            \
        false, a, false, bs##s, (short)0, acc8, false, false);               \
  }

__global__ __launch_bounds__(256) void psl_trans_kernel(
    const float* __restrict__ dist, const long long* __restrict__ hypo_len,
    const int* __restrict__ comp, float* __restrict__ partial) {
  __shared__ float rowbuf[LDSF] __attribute__((aligned(16)));
  __shared__ float wsum[8];

  const int b = blockIdx.x;
  const int tid = threadIdx.x;
  const int wave = tid >> 5;
  const int lane = tid & 31;
  const bool hsel = lane >= 16;  // which 16-lane half of the wave
  const int m = lane & 15;       // A row / D column within the 16x16 tile

  // ---- Stage the whole 14KB row HBM -> LDS with async b128 DMA loads ----
  const unsigned lds0 = (unsigned)(uintptr_t)(void*)rowbuf;
  const float* srow = dist + (size_t)b * ROWF;
  for (int c = tid; c < CHUNKS; c += 256) {
    const unsigned ldsa = lds0 + ((unsigned)c << 4);
    const unsigned goff = (unsigned)c << 4;
    asm volatile("global_load_async_to_lds_b128 %0, %1, %2"
                 :
                 : "v"(ldsa), "v"(goff), "s"(srow)
                 : "memory");
  }
  asm volatile("s_wait_asynccnt 0" ::: "memory");
  __syncthreads();

  const int hlen = (int)hypo_len[b];

  // ---- Per-block constant B operands (13 x v2f, live in VGPRs) ----
  FOR13(BDEF)

  float accs = 0.0f;

  // 8 waves x 4 tiles cover 32 tiles of 16 positions (510 valid t, rest masked)
  for (int it = 0; it < 4; ++it) {
    const int t0 = (wave + (it << 3)) << 4;
    const int t = t0 + m;  // this lane's row of the A tile

    // p_t (7) and p_t1 (7) are 14 contiguous LDS floats at t*7
    float pq[14];
    const int pbase = t * 7;
#pragma unroll
    for (int ii = 0; ii < 14; ++ii) pq[ii] = rowbuf[pbase + ii];

    v8f acc8 = {0.f, 0.f, 0.f, 0.f, 0.f, 0.f, 0.f, 0.f};
    FOR13(WSTEP)  // composed tile: D[row][k] for 16 positions

    // D layout: lane L, reg v -> row = v + 8*(L/16), col k = L%16
    const int h8 = hsel ? 8 : 0;
#pragma unroll
    for (int v = 0; v < 8; ++v) {
      const int row = t0 + v + h8;
      const float p2 = (m < NREL) ? rowbuf[(row + 2) * NREL + m] : 0.0f;
      float r = acc8[v] - p2;
      r = r > 0.0f ? r : 0.0f;                        // relu
      accs += ((row + 2) < hlen) ? r : 0.0f;          // mask (also kills pads)
    }
  }

  // ---- Deterministic block reduction ----
  float a = accs;
#pragma unroll
  for (int off = 16; off > 0; off >>= 1) a += __shfl_xor(a, off, 32);
  if (lane == 0) wsum[wave] = a;
  __syncthreads();
  if (tid == 0) {
    float s = 0.0f;
#pragma unroll
    for (int wv = 0; wv < 8; ++wv) s += wsum[wv];
    partial[b] = s;
  }
}

__global__ __launch_bounds__(256) void psl_finalize_kernel(
    const float* __restrict__ partial, const long long* __restrict__ hypo_len,
    float* __restrict__ out, int batch) {
  __shared__ float s_num[256];
  __shared__ float s_den[256];
  const int tid = threadIdx.x;
  float num = 0.0f, den = 0.0f;
  for (int i = tid; i < batch; i += 256) {
    num += partial[i];
    int c = (int)hypo_len[i] - 2;
    c = c < 0 ? 0 : (c > (MLEN - 2) ? (MLEN - 2) : c);
    den += (float)c;
  }
  s_num[tid] = num;
  s_den[tid] = den;
  __syncthreads();
  for (int off = 128; off > 0; off >>= 1) {
    if (tid < off) {
      s_num[tid] += s_num[tid + off];
      s_den[tid] += s_den[tid + off];
    }
    __syncthreads();
  }
  if (tid == 0) out[0] = 0.7f * (s_num[0] / (s_den[0] + 1e-8f));
}

extern "C" void kernel_launch(void* const* d_in, const int* in_sizes, int n_in,
                              void* d_out, int out_size, void* d_ws,
                              size_t ws_size, hipStream_t stream) {
  (void)n_in;
  (void)out_size;
  (void)ws_size;
  const float* dist = (const float*)d_in[0];
  const long long* hlen = (const long long*)d_in[1];
  const int* comp = (const int*)d_in[2];
  float* out = (float*)d_out;
  float* partial = (float*)d_ws;  // batch floats of per-row partial sums
  const int batch = in_sizes[1];  // 8192

  psl_trans_kernel<<<batch, 256, 0, stream>>>(dist, hlen, comp, partial);
  psl_finalize_kernel<<<1, 256, 0, stream>>>(partial, hlen, out, batch);
}